// Relational_GCN_8589935118
// MI455X (gfx1250) — compile-verified
//
#include <hip/hip_runtime.h>
#include <hip/hip_bf16.h>

typedef __attribute__((ext_vector_type(2))) float v2f;
typedef __attribute__((ext_vector_type(8))) float v8f;

#define N_NODES 50000
#define N_EDGES 800000
#define DIM     64
#define N_RELS  8
#define N_BASES 4

// ---------------------------------------------------------------------------
// W[r,i,o] = sum_b coeff[r,b] * basis[b,i,o]   (8*64*64 outputs)
// ---------------------------------------------------------------------------
__global__ __launch_bounds__(256)
void rgcn_compute_w(float* __restrict__ W, const float* __restrict__ coeff,
                    const float* __restrict__ basis) {
    int idx = blockIdx.x * 256 + threadIdx.x;      // < 8*4096
    int r   = idx >> 12;
    int io  = idx & 4095;
    float acc = 0.f;
#pragma unroll
    for (int b = 0; b < N_BASES; ++b)
        acc += coeff[r * N_BASES + b] * basis[b * 4096 + io];
    W[idx] = acc;
}

// ---------------------------------------------------------------------------
// C[nrows,64] = A[nrows,64] @ B[64,64] (+ bias) via V_WMMA_F32_16X16X4_F32.
// One wave computes a 16x64 tile: 4 accumulators (16x16), 16 K-steps of 4.
// A frag (ISA 7.12.2, 32-bit A 16x4): lane&15 = M, K = 4*kk + 2*half + {0,1}.
// B frag symmetric: lane&15 = N, same K split.
// C/D: VGPR i -> M = i + 8*half, N = lane&15.
//
// B is staged in LDS in K-PAIR-INTERLEAVED layout:
//   Bs[(k>>1)*128 + 2*n + (k&1)] = B[k][n]
// so each B fragment is ONE aligned ds_load_b64 landing in an adjacent VGPR
// pair (no v_dual_mov repacking, conflict-free banks on read).
// ---------------------------------------------------------------------------
__global__ __launch_bounds__(256)
void rgcn_gemm64(const float* __restrict__ A, const float* __restrict__ B,
                 const float* __restrict__ bias, float* __restrict__ C,
                 int nrows) {
    __shared__ float Bs[DIM * DIM];                // 16 KB, pair-interleaved
    // cooperative pack: reads coalesced, one-time 16 KB shuffle
#pragma unroll
    for (int j = 0; j < 16; ++j) {
        int idx = threadIdx.x + j * 256;           // 4096 elements
        int k   = idx >> 6;
        int n   = idx & 63;
        Bs[(k >> 1) * 128 + 2 * n + (k & 1)] = B[idx];
    }
    __syncthreads();

    int tile   = blockIdx.x * 8 + (threadIdx.x >> 5);
    int ntiles = nrows >> 4;
    if (tile >= ntiles) return;                    // wave-uniform exit

    int lane = threadIdx.x & 31;
    int m    = lane & 15;
    int half = lane >> 4;

    const float* arow = A + (size_t)(tile * 16 + m) * DIM;

    v8f acc0 = {}, acc1 = {}, acc2 = {}, acc3 = {};
#pragma unroll
    for (int kk = 0; kk < 16; ++kk) {
        int kb = kk * 4 + half * 2;                // this lane's K pair
        v2f a = *(const v2f*)(arow + kb);          // A frag: 8B aligned b64

        const float* bp = Bs + (kk * 2 + half) * 128;   // pair row (kb>>1)
        v2f b0 = *(const v2f*)(bp + 2 * m);
        v2f b1 = *(const v2f*)(bp + 2 * (16 + m));
        v2f b2 = *(const v2f*)(bp + 2 * (32 + m));
        v2f b3 = *(const v2f*)(bp + 2 * (48 + m));
        acc0 = __builtin_amdgcn_wmma_f32_16x16x4_f32(false, a, false, b0, (short)0, acc0, false, false);
        acc1 = __builtin_amdgcn_wmma_f32_16x16x4_f32(false, a, false, b1, (short)0, acc1, false, false);
        acc2 = __builtin_amdgcn_wmma_f32_16x16x4_f32(false, a, false, b2, (short)0, acc2, false, false);
        acc3 = __builtin_amdgcn_wmma_f32_16x16x4_f32(false, a, false, b3, (short)0, acc3, false, false);
    }

    float bv0 = 0.f, bv1 = 0.f, bv2 = 0.f, bv3 = 0.f;
    if (bias) { bv0 = bias[m]; bv1 = bias[16 + m]; bv2 = bias[32 + m]; bv3 = bias[48 + m]; }

    int rbase = tile * 16 + half * 8;
#pragma unroll
    for (int i = 0; i < 8; ++i) {
        float* crow = C + (size_t)(rbase + i) * DIM;
        crow[m]      = acc0[i] + bv0;
        crow[16 + m] = acc1[i] + bv1;
        crow[32 + m] = acc2[i] + bv2;
        crow[48 + m] = acc3[i] + bv3;
    }
}

// ---------------------------------------------------------------------------
// For edges with etype==rel: Hout[dst] += XWr[src]   (one wave per edge,
// 2 columns per lane, f32 global atomics). XWr/Hout are L2-resident.
// ---------------------------------------------------------------------------
__global__ __launch_bounds__(256)
void rgcn_scatter(float* __restrict__ Hout, const float* __restrict__ XWr,
                  const int* __restrict__ src, const int* __restrict__ dst,
                  const int* __restrict__ etype, int rel, int nedges) {
    int e = blockIdx.x * 8 + (threadIdx.x >> 5);   // wave-uniform
    if (e >= nedges) return;
    if (etype[e] != rel) return;
    int lane = threadIdx.x & 31;
    int s = src[e], d = dst[e];
    v2f mv = *(const v2f*)(XWr + (size_t)s * DIM + lane * 2);
    float* orow = Hout + (size_t)d * DIM + lane * 2;
    atomicAdd(orow,     mv.x);
    atomicAdd(orow + 1, mv.y);
}

// ---------------------------------------------------------------------------
// leaky_relu, slope 0.01 (in place)
// ---------------------------------------------------------------------------
__global__ __launch_bounds__(256)
void rgcn_leaky(float* __restrict__ h, int n) {
    int i = blockIdx.x * 256 + threadIdx.x;
    if (i < n) {
        float v = h[i];
        h[i] = v > 0.f ? v : 0.01f * v;
    }
}

extern "C" void kernel_launch(void* const* d_in, const int* in_sizes, int n_in,
                              void* d_out, int out_size, void* d_ws, size_t ws_size,
                              hipStream_t stream) {
    const float* x      = (const float*)d_in[0];
    const int*   src    = (const int*)  d_in[1];
    const int*   dst    = (const int*)  d_in[2];
    const int*   etype  = (const int*)  d_in[3];
    const float* basis1 = (const float*)d_in[4];
    const float* coeff1 = (const float*)d_in[5];
    const float* loopw1 = (const float*)d_in[6];
    const float* bias1  = (const float*)d_in[7];
    const float* basis2 = (const float*)d_in[8];
    const float* coeff2 = (const float*)d_in[9];
    const float* loopw2 = (const float*)d_in[10];
    const float* bias2  = (const float*)d_in[11];
    float* out = (float*)d_out;

    // workspace layout (all fully written before read): ~25.7 MB total
    float* W   = (float*)d_ws;                     // 8*64*64        (128 KB)
    float* XWr = W + N_RELS * DIM * DIM;           // N_NODES*64     (12.8 MB)
    float* H1  = XWr + (size_t)N_NODES * DIM;      // N_NODES*64     (12.8 MB)

    const int wBlocks    = (N_RELS * DIM * DIM) / 256;       // 128
    const int gemmBlocks = ((N_NODES / 16) + 7) / 8;         // 391 (3125 tiles)
    const int scatBlocks = (N_EDGES + 7) / 8;                // 100000
    const int eltBlocks  = (N_NODES * DIM + 255) / 256;

    // ---- Layer 1: H1 = segsum(XW[etype,src]->dst) + x@loopw1 + bias1 ----
    rgcn_compute_w<<<wBlocks, 256, 0, stream>>>(W, coeff1, basis1);
    rgcn_gemm64<<<gemmBlocks, 256, 0, stream>>>(x, loopw1, bias1, H1, N_NODES);
    for (int r = 0; r < N_RELS; ++r) {
        rgcn_gemm64<<<gemmBlocks, 256, 0, stream>>>(x, W + r * DIM * DIM, nullptr, XWr, N_NODES);
        rgcn_scatter<<<scatBlocks, 256, 0, stream>>>(H1, XWr, src, dst, etype, r, N_EDGES);
    }
    rgcn_leaky<<<eltBlocks, 256, 0, stream>>>(H1, N_NODES * DIM);

    // ---- Layer 2: out = segsum + H1@loopw2 + bias2 ----
    rgcn_compute_w<<<wBlocks, 256, 0, stream>>>(W, coeff2, basis2);
    rgcn_gemm64<<<gemmBlocks, 256, 0, stream>>>(H1, loopw2, bias2, out, N_NODES);
    for (int r = 0; r < N_RELS; ++r) {
        rgcn_gemm64<<<gemmBlocks, 256, 0, stream>>>(H1, W + r * DIM * DIM, nullptr, XWr, N_NODES);
        rgcn_scatter<<<scatBlocks, 256, 0, stream>>>(out, XWr, src, dst, etype, r, N_EDGES);
    }
}